// WaveletLinear_79517024518853
// MI455X (gfx1250) — compile-verified
//
#include <hip/hip_runtime.h>
#include <math.h>
#include <stdint.h>
#include <stddef.h>

// ---------------- tile configuration ----------------
// Block: 256 threads (8 wave32s). Output tile: BT x OT = 64 x 32.
// Reduction over I in chunks of KT = 32, double-buffered in LDS with
// CDNA5 async global->LDS copies (ASYNCcnt).
constexpr int BT = 64;        // batch-rows per block
constexpr int OT = 32;        // out-features per block
constexpr int KT = 32;        // k-chunk (in-features per stage)
constexpr int KP = KT + 4;    // padded x row: stride 144B -> conflict-free
constexpr int NTHREADS = 256;

#define A_MIN_F 0.001f
#define EPS_F   1e-8f
// exp(-0.5*s2) == exp2(s2 * (-0.5*log2(e)))
#define NEG_HALF_LOG2E (-0.7213475204444817f)

typedef int   v4i_t __attribute__((ext_vector_type(4)));
typedef float v2f   __attribute__((ext_vector_type(2)));
typedef __attribute__((address_space(1))) v4i_t g_v4i;   // global int4
typedef __attribute__((address_space(3))) v4i_t l_v4i;   // LDS int4

// ---- CDNA5 async global -> LDS copy (16B per lane), tracked by ASYNCcnt ----
__device__ __forceinline__ void async_copy_b128(const void* g, void* l) {
#if __has_builtin(__builtin_amdgcn_global_load_async_to_lds_b128)
  __builtin_amdgcn_global_load_async_to_lds_b128(
      (g_v4i*)g, (l_v4i*)l, /*offset=*/0, /*cpol=*/0);
#else
  unsigned loff = (unsigned)(size_t)(l_v4i*)l;            // LDS byte offset
  unsigned long long ga = (unsigned long long)(size_t)g;  // 64-bit global
  asm volatile("global_load_async_to_lds_b128 %0, %1, off"
               :: "v"(loff), "v"(ga) : "memory");
#endif
}

template <int N>
__device__ __forceinline__ void wait_asynccnt() {
#if __has_builtin(__builtin_amdgcn_s_wait_asynccnt)
  __builtin_amdgcn_s_wait_asynccnt(N);
#else
  asm volatile("s_wait_asynccnt %0" :: "i"(N) : "memory");
#endif
}

#if __has_builtin(__builtin_amdgcn_exp2f)
#define FAST_EXP2(x) __builtin_amdgcn_exp2f(x)   // v_exp_f32
#else
#define FAST_EXP2(x) __exp2f(x)
#endif

__global__ __launch_bounds__(NTHREADS)
void wavelet_linear_kernel(const float* __restrict__ X,   // (B, I)
                           const float* __restrict__ T,   // (O, I)
                           const float* __restrict__ S,   // (O, I) scale_raw
                           const float* __restrict__ W,   // (O, I)
                           float* __restrict__ Y,         // (B, O)
                           int B, int O, int I) {
  __shared__ float xs[2][BT][KP];   // x tile, k-contiguous, padded rows
  __shared__ float pt[2][KT][OT];   // translation,              [k][o]
  __shared__ float pc[2][KT][OT];   // inv2 * (-0.5*log2e),      [k][o]
  __shared__ float pw[2][KT][OT];   // w,                        [k][o]
  __shared__ float pu[2][KT][OT];   // w * inv2,                 [k][o]

  const int tid = (int)threadIdx.x;
  const int o0  = (int)blockIdx.x * OT;
  const int b0  = (int)blockIdx.y * BT;
  const int nstages = I / KT;

  // param staging map: one float4 per thread per array
  const int po  = tid >> 3;         // o-row within tile [0,32)
  const int pk4 = (tid & 7) * 4;    // k start (float4)  [0,32)

  // compute map: 4 b-rows x 2 o-cols per thread
  const int tx = tid & 15;          // -> o
  const int ty = tid >> 4;          // -> b
  const int bl = ty * 4;
  const int ol = tx * 2;

  // Issue async x-tile copies for stage s into buffer `buf`.
  // BT*KT floats = 512 x 16B chunks => exactly 2 async b128 per thread.
  auto issue_x = [&](int s, int buf) {
    const int k0 = s * KT;
#pragma unroll
    for (int u = 0; u < 2; ++u) {
      int c   = tid + u * NTHREADS;     // chunk id [0,512)
      int row = c >> 3;                 // 8 chunks per 32-float row
      int col = (c & 7) * 4;            // float index within row
      async_copy_b128(&X[(size_t)(b0 + row) * I + k0 + col],
                      &xs[buf][row][col]);
    }
  };

  float4 rT, rW, rS;
  auto load_params = [&](int s) {   // global -> regs (prefetch)
    const size_t base = (size_t)(o0 + po) * I + (size_t)s * KT + pk4;
    rT = *reinterpret_cast<const float4*>(T + base);
    rW = *reinterpret_cast<const float4*>(W + base);
    rS = *reinterpret_cast<const float4*>(S + base);
  };
  // regs -> LDS: transpose + fold scale into exp-coef and w*inv2.
  // Cold path: softplus/rcp amortized over BT=64 hot uses per element.
  auto store_params = [&](int buf) {
    const float tv[4] = {rT.x, rT.y, rT.z, rT.w};
    const float wv[4] = {rW.x, rW.y, rW.z, rW.w};
    const float sv[4] = {rS.x, rS.y, rS.z, rS.w};
#pragma unroll
    for (int j = 0; j < 4; ++j) {
      const float z    = sv[j];
      const float sp   = fmaxf(z, 0.0f) + log1pf(expf(-fabsf(z)));
      const float sc   = A_MIN_F + sp + EPS_F;
      const float inv  = 1.0f / sc;
      const float inv2 = inv * inv;
      pt[buf][pk4 + j][po] = tv[j];
      pc[buf][pk4 + j][po] = inv2 * NEG_HALF_LOG2E;
      pw[buf][pk4 + j][po] = wv[j];
      pu[buf][pk4 + j][po] = wv[j] * inv2;
    }
  };

  v2f acc[4];
#pragma unroll
  for (int j = 0; j < 4; ++j) acc[j] = (v2f){0.f, 0.f};

  // Hot loop: packed-f32 (V_PK_*) math over the o-pair; v_exp_f32 on the
  // trans pipe co-executes with the VALU stream.
  // Per o-pair: 5 pk ops + 2 v_exp_f32.
  auto compute = [&](int buf) {
#pragma unroll
    for (int kk = 0; kk < KT; kk += 4) {
      float4 xq[4];                               // 1 ds_load_b128 per b-row
#pragma unroll
      for (int j = 0; j < 4; ++j)
        xq[j] = *reinterpret_cast<const float4*>(&xs[buf][bl + j][kk]);
      v2f tv[4], cv[4], wv[4], uv[4];
#pragma unroll
      for (int q = 0; q < 4; ++q) {
        tv[q] = *reinterpret_cast<const v2f*>(&pt[buf][kk + q][ol]);
        cv[q] = *reinterpret_cast<const v2f*>(&pc[buf][kk + q][ol]);
        wv[q] = *reinterpret_cast<const v2f*>(&pw[buf][kk + q][ol]);
        uv[q] = *reinterpret_cast<const v2f*>(&pu[buf][kk + q][ol]);
      }
#pragma unroll
      for (int q = 0; q < 4; ++q) {
#pragma unroll
        for (int j = 0; j < 4; ++j) {
          const float xj = (q == 0) ? xq[j].x
                         : (q == 1) ? xq[j].y
                         : (q == 2) ? xq[j].z : xq[j].w;
          v2f xsp = (v2f){xj, xj};
          v2f d   = xsp - tv[q];                  // v_pk_add (neg)
          v2f dd  = d * d;                        // v_pk_mul
          v2f a   = dd * cv[q];                   // v_pk_mul  (= -0.5*s2*log2e)
          v2f e;
          e.x = FAST_EXP2(a.x);                   // v_exp_f32 (trans)
          e.y = FAST_EXP2(a.y);
          v2f r = __builtin_elementwise_fma(-uv[q], dd, wv[q]);  // w*(1-s2)
          acc[j] = __builtin_elementwise_fma(r, e, acc[j]);      // v_pk_fma
        }
      }
    }
  };

  // ---------------- software pipeline ----------------
  issue_x(0, 0);
  load_params(0);
  for (int s = 0; s < nstages; ++s) {
    const int buf = s & 1;
    const bool more = (s + 1) < nstages;
    if (more) issue_x(s + 1, buf ^ 1);   // prefetch next x tile (async)
    store_params(buf);                   // transform+transpose current params
    if (more) load_params(s + 1);        // prefetch next params into regs
    if (more) wait_asynccnt<2>();        // this stage's 2 copies done (in-order)
    else      wait_asynccnt<0>();
    __syncthreads();                     // publish LDS (async + ds_store)
    compute(buf);
    __syncthreads();                     // release buf before it is re-filled
  }

  // ---------------- epilogue: write 4x2 outputs ----------------
#pragma unroll
  for (int j = 0; j < 4; ++j) {
    *reinterpret_cast<v2f*>(&Y[(size_t)(b0 + bl + j) * O + o0 + ol]) = acc[j];
  }
}

extern "C" void kernel_launch(void* const* d_in, const int* in_sizes, int n_in,
                              void* d_out, int out_size, void* d_ws, size_t ws_size,
                              hipStream_t stream) {
  (void)n_in; (void)d_ws; (void)ws_size;
  const float* x  = (const float*)d_in[0];
  const float* tr = (const float*)d_in[1];
  const float* sr = (const float*)d_in[2];
  const float* w  = (const float*)d_in[3];
  float* y = (float*)d_out;

  // Recover dims: x=(B,I), params=(O,I), out=(B,O)
  const double sx = (double)in_sizes[0];   // B*I
  const double sp = (double)in_sizes[1];   // O*I
  const double so = (double)out_size;      // B*O
  const int I = (int)(sqrt(sx * sp / so) + 0.5);
  const int B = (int)(sx / I + 0.5);
  const int O = (int)(sp / I + 0.5);

  dim3 grid((unsigned)(O / OT), (unsigned)(B / BT));
  dim3 block(NTHREADS);
  wavelet_linear_kernel<<<grid, block, 0, stream>>>(x, tr, sr, w, y, B, O, I);
}